// Model_42434276884991
// MI455X (gfx1250) — compile-verified
//
#include <hip/hip_runtime.h>

// ---- problem constants (match reference) ----
#define N_NODES  100000
#define N_EDGES  3200000
#define D_IN     96
#define D_OUT    32
#define N_HEADS  4
#define LEAKY    0.2f
#define OUT_COLS (N_HEADS * D_OUT)          // 128
#define OUT_TOT  (N_NODES * OUT_COLS)       // 12.8M floats

typedef float v2f __attribute__((ext_vector_type(2)));
typedef float v8f __attribute__((ext_vector_type(8)));

// ------------------------------------------------------------------
// Phase 0: zero the output accumulator (scatter-add target).
// ------------------------------------------------------------------
__global__ void zero_out_kernel(float4* __restrict__ out) {
    int i = blockIdx.x * blockDim.x + threadIdx.x;   // OUT_TOT/4 = 3.2M
    if (i < OUT_TOT / 4) out[i] = make_float4(0.f, 0.f, 0.f, 0.f);
}

// ------------------------------------------------------------------
// Phase 1: h[head][node][0:32] = x @ W[head] via V_WMMA_F32_16X16X4_F32.
// grid.x = 6250 M-tiles (100000/16 exact), block = 256 = 8 waves.
// wave -> (head, n-tile of 16), each wave produces one 16x16 tile.
// A fragment (16x4 f32): lanes 0-15 hold K=k,k+1; lanes 16-31 hold K=k+2,k+3.
// B fragment (4x16 f32): mirrored row-striping across lanes.
// D fragment: VGPR r holds row (r + 8*half), col = lane&15.
// ------------------------------------------------------------------
__global__ void gemm_wmma_kernel(const float* __restrict__ x,
                                 const float* __restrict__ W,
                                 float* __restrict__ h) {
    const int wave  = threadIdx.x >> 5;
    const int lane  = threadIdx.x & 31;
    const int head  = wave >> 1;
    const int ntile = wave & 1;
    const int mbase = blockIdx.x * 16;
    const int nbase = ntile * 16;
    const int half  = lane >> 4;      // which 16-lane half
    const int mn    = lane & 15;      // row for A, col for B/D

    const float* __restrict__ Wh = W + (size_t)head * D_IN * D_OUT;
    const float* __restrict__ xr = x + (size_t)(mbase + mn) * D_IN;

    v8f c = {0.f, 0.f, 0.f, 0.f, 0.f, 0.f, 0.f, 0.f};

    #pragma unroll
    for (int k0 = 0; k0 < D_IN; k0 += 4) {
        const int ka = k0 + 2 * half;
        v2f a, b;
        a.x = xr[ka];
        a.y = xr[ka + 1];
        b.x = Wh[(size_t)ka * D_OUT + nbase + mn];
        b.y = Wh[(size_t)(ka + 1) * D_OUT + nbase + mn];
        c = __builtin_amdgcn_wmma_f32_16x16x4_f32(
                /*neg_a=*/false, a, /*neg_b=*/false, b,
                /*c_mod=*/(short)0, c, /*reuse_a=*/false, /*reuse_b=*/false);
    }

    float* __restrict__ hh = h + (size_t)head * N_NODES * D_OUT;
    #pragma unroll
    for (int r = 0; r < 8; ++r) {
        const int m = mbase + r + 8 * half;
        hh[(size_t)m * D_OUT + nbase + mn] = c[r];
    }
}

// ------------------------------------------------------------------
// Phase 2: per-node attention projections
//   alpha1[head*N+n] = h[head][n] . a[head, :32]
//   alpha2[head*N+n] = h[head][n] . a[head, 32:]
// (makes the per-edge score two scalar gathers instead of a 64-dot)
// ------------------------------------------------------------------
__global__ void alpha_kernel(const float* __restrict__ h,
                             const float* __restrict__ a,
                             float* __restrict__ alpha1,
                             float* __restrict__ alpha2) {
    int idx = blockIdx.x * blockDim.x + threadIdx.x;   // head*N + node
    if (idx >= N_HEADS * N_NODES) return;
    const int head = idx / N_NODES;
    const float4* __restrict__ hr = (const float4*)(h + (size_t)idx * D_OUT);
    const float4* __restrict__ a1 = (const float4*)(a + (size_t)head * 2 * D_OUT);
    const float4* __restrict__ a2 = a1 + (D_OUT / 4);
    float s1 = 0.f, s2 = 0.f;
    #pragma unroll
    for (int j = 0; j < D_OUT / 4; ++j) {
        float4 hv = hr[j], v1 = a1[j], v2 = a2[j];
        s1 += hv.x * v1.x + hv.y * v1.y + hv.z * v1.z + hv.w * v1.w;
        s2 += hv.x * v2.x + hv.y * v2.y + hv.z * v2.z + hv.w * v2.w;
    }
    alpha1[idx] = s1;
    alpha2[idx] = s2;
}

// ------------------------------------------------------------------
// Phase 3: edge scatter. Thread = (head, edge); head = idx / E so that
// edge_src/edge_dst loads coalesce within a wave. Gathers h[head][dst]
// (128 B row, L2-resident) and fires 32 no-return f32 atomics into the
// L2-resident output accumulator.
// ------------------------------------------------------------------
__global__ void scatter_kernel(const float* __restrict__ h,
                               const float* __restrict__ alpha1,
                               const float* __restrict__ alpha2,
                               const int* __restrict__ esrc,
                               const int* __restrict__ edst,
                               float* __restrict__ out) {
    int idx = blockIdx.x * blockDim.x + threadIdx.x;   // N_HEADS * N_EDGES
    if (idx >= N_HEADS * N_EDGES) return;
    const int head = idx / N_EDGES;
    const int e    = idx - head * N_EDGES;

    const int s = esrc[e];
    const int d = edst[e];

    const float sc = alpha1[head * N_NODES + s] + alpha2[head * N_NODES + d];
    const float lr = sc > 0.f ? sc : LEAKY * sc;
    const float w  = __expf(-lr);

    const float4* __restrict__ hrow =
        (const float4*)(h + ((size_t)head * N_NODES + d) * D_OUT);
    float* __restrict__ orow = out + (size_t)s * OUT_COLS + head * D_OUT;

    #pragma unroll
    for (int j = 0; j < D_OUT / 4; ++j) {
        float4 v = hrow[j];
        atomicAdd(orow + j * 4 + 0, w * v.x);
        atomicAdd(orow + j * 4 + 1, w * v.y);
        atomicAdd(orow + j * 4 + 2, w * v.z);
        atomicAdd(orow + j * 4 + 3, w * v.w);
    }
}

// ------------------------------------------------------------------
// Phase 4: ELU in place on the accumulated output.
// ------------------------------------------------------------------
__global__ void elu_kernel(float* __restrict__ out) {
    int i = blockIdx.x * blockDim.x + threadIdx.x;
    if (i < OUT_TOT) {
        float v = out[i];
        out[i] = v > 0.f ? v : (__expf(v) - 1.f);
    }
}

// ------------------------------------------------------------------
// Launch: inputs are (x, W, a, edge_src, edge_dst) per setup_inputs().
// Workspace layout: h (4*100000*32 f32 = 51.2MB) | alpha1 (1.6MB) | alpha2.
// Scatter-adds go straight into d_out (zeroed here each call -> graph-safe).
// ------------------------------------------------------------------
extern "C" void kernel_launch(void* const* d_in, const int* in_sizes, int n_in,
                              void* d_out, int out_size, void* d_ws, size_t ws_size,
                              hipStream_t stream) {
    const float* x    = (const float*)d_in[0];
    const float* W    = (const float*)d_in[1];
    const float* a    = (const float*)d_in[2];
    const int*   esrc = (const int*)d_in[3];
    const int*   edst = (const int*)d_in[4];
    float*       out  = (float*)d_out;

    float* h      = (float*)d_ws;                                  // [4][100000][32]
    float* alpha1 = h + (size_t)N_HEADS * N_NODES * D_OUT;         // [4][100000]
    float* alpha2 = alpha1 + (size_t)N_HEADS * N_NODES;            // [4][100000]

    // 0) zero output accumulator
    {
        int n = OUT_TOT / 4;
        zero_out_kernel<<<(n + 255) / 256, 256, 0, stream>>>((float4*)out);
    }
    // 1) h = x @ W per head (WMMA f32 16x16x4)
    gemm_wmma_kernel<<<N_NODES / 16, 256, 0, stream>>>(x, W, h);
    // 2) per-node alphas
    {
        int n = N_HEADS * N_NODES;
        alpha_kernel<<<(n + 255) / 256, 256, 0, stream>>>(h, a, alpha1, alpha2);
    }
    // 3) edge scatter-add
    {
        int n = N_HEADS * N_EDGES;
        scatter_kernel<<<(n + 255) / 256, 256, 0, stream>>>(h, alpha1, alpha2,
                                                            esrc, edst, out);
    }
    // 4) ELU in place
    elu_kernel<<<(OUT_TOT + 255) / 256, 256, 0, stream>>>(out);
}